// GCNEncoder_6786048328254
// MI455X (gfx1250) — compile-verified
//
#include <hip/hip_runtime.h>

typedef __attribute__((ext_vector_type(2))) float v2f;
typedef __attribute__((ext_vector_type(8))) float v8f;

#define N_NODES  100000
#define N_EDGES  1600000
#define IN_DIM   128
#define HID_DIM  64
#define EMB_DIM  128
#define N_GRAPHS 256
#define BN_EPS   1e-5f

// ---------------------------------------------------------------- utilities

__global__ void zero_f(float* __restrict__ p, int n) {
  int i = blockIdx.x * blockDim.x + threadIdx.x;
  if (i < n) p[i] = 0.0f;
}

template <int D>
__global__ void init_bias(float* __restrict__ agg, const float* __restrict__ b) {
  long long i = (long long)blockIdx.x * blockDim.x + threadIdx.x;
  if (i < (long long)N_NODES * D) agg[i] = b[(int)(i & (D - 1))];
}

// ------------------------------------------------------------ WMMA f32 GEMM
// C[M,N] = A[M,K] * B[K,N], fp32, V_WMMA_F32_16X16X4_F32.
// grid.x = M/16 (row tile), blockDim.x = (N/16)*32: one wave per 16-col tile.
// A 16x4 layout: lane holds A[row = lane&15][k = (lane>>4)*2 + {0,1}]
// B 4x16 layout: lane holds B[k = (lane>>4)*2 + {0,1}][col = lane&15]
// C/D 16x16:     vgpr i holds C[row = i + (lane>>4)*8][col = lane&15]
__global__ void gemm_wmma_f32(const float* __restrict__ A,
                              const float* __restrict__ B,
                              float* __restrict__ C,
                              int N, int K) {
  const int wave = threadIdx.x >> 5;
  const int lane = threadIdx.x & 31;
  const int m0   = blockIdx.x << 4;
  const int n0   = wave << 4;
  const int r    = lane & 15;
  const int kg   = (lane >> 4) << 1;   // 0 or 2

  const float* __restrict__ arow = A + (long long)(m0 + r) * K;

  v8f acc = {0.f, 0.f, 0.f, 0.f, 0.f, 0.f, 0.f, 0.f};
  for (int k0 = 0; k0 < K; k0 += 4) {
    v2f a, b;
    a.x = arow[k0 + kg];
    a.y = arow[k0 + kg + 1];
    b.x = B[(long long)(k0 + kg) * N + n0 + r];
    b.y = B[(long long)(k0 + kg + 1) * N + n0 + r];
    acc = __builtin_amdgcn_wmma_f32_16x16x4_f32(
        /*neg_a=*/false, a, /*neg_b=*/false, b,
        /*c_mod=*/(short)0, acc, /*reuse_a=*/false, /*reuse_b=*/false);
  }

  const int crow = m0 + ((lane >> 4) << 3);
  const int ccol = n0 + r;
#pragma unroll
  for (int i = 0; i < 8; ++i)
    C[(long long)(crow + i) * N + ccol] = acc[i];
}

// ----------------------------------------------------- edge-weighted scatter
// D/4 threads per edge; float4 gather from h[src], 4 float atomics to agg[dst].
template <int D>
__global__ void scatter_edges(const float* __restrict__ h,
                              float* __restrict__ agg,
                              const int* __restrict__ src,
                              const int* __restrict__ dst,
                              const float* __restrict__ ew) {
  const int chunkShift = (D == 64) ? 4 : 5;        // D/4 chunks per edge
  const long long total = (long long)N_EDGES << chunkShift;
  long long t = (long long)blockIdx.x * blockDim.x + threadIdx.x;
  if (t >= total) return;
  const int e = (int)(t >> chunkShift);
  const int c = (int)(t & ((1 << chunkShift) - 1)) << 2;
  const int s = src[e];
  const int d = dst[e];
  const float w = ew[e];
  const float4 v = *(const float4*)(h + (long long)s * D + c);
  float* __restrict__ o = agg + (long long)d * D + c;
  atomicAdd(o + 0, v.x * w);
  atomicAdd(o + 1, v.y * w);
  atomicAdd(o + 2, v.z * w);
  atomicAdd(o + 3, v.w * w);
}

// ------------------------------------------------------------ BN statistics
template <int D>
__global__ void col_stats(const float* __restrict__ v,
                          float* __restrict__ sum,
                          float* __restrict__ sqsum) {
  __shared__ float shs[256];
  __shared__ float shq[256];
  const int tid = threadIdx.x;
  const int c   = tid & (D - 1);
  const int rpb = 256 / D;                          // rows per iteration
  float s = 0.f, q = 0.f;
  for (long long row = (long long)blockIdx.x * rpb + (tid / D);
       row < N_NODES; row += (long long)gridDim.x * rpb) {
    float x = v[row * D + c];
    s += x;
    q += x * x;
  }
  shs[tid] = s;
  shq[tid] = q;
  __syncthreads();
  if (tid < D) {
    for (int k = tid + D; k < 256; k += D) { s += shs[k]; q += shq[k]; }
    atomicAdd(&sum[c], s);
    atomicAdd(&sqsum[c], q);
  }
}

__global__ void finalize_stats(const float* __restrict__ sum,
                               const float* __restrict__ sqsum,
                               float* __restrict__ mean,
                               float* __restrict__ rstd, int D) {
  int c = blockIdx.x * blockDim.x + threadIdx.x;
  if (c < D) {
    const float invN = 1.0f / (float)N_NODES;
    float m   = sum[c] * invN;
    float var = fmaxf(sqsum[c] * invN - m * m, 0.0f);
    mean[c] = m;
    rstd[c] = rsqrtf(var + BN_EPS);
  }
}

template <int D>
__global__ void bn_relu(const float* __restrict__ v, float* __restrict__ out,
                        const float* __restrict__ mean,
                        const float* __restrict__ rstd,
                        const float* __restrict__ g,
                        const float* __restrict__ be) {
  long long i = (long long)blockIdx.x * blockDim.x + threadIdx.x;
  if (i >= (long long)N_NODES * D) return;
  const int c = (int)(i & (D - 1));
  float y = (v[i] - mean[c]) * rstd[c] * g[c] + be[c];
  out[i] = y > 0.0f ? y : 0.0f;
}

// ----------------------------------------------------------------- pooling
__global__ void pool_count(const int* __restrict__ bv, float* __restrict__ cnt) {
  int i = blockIdx.x * blockDim.x + threadIdx.x;
  if (i < N_NODES) atomicAdd(&cnt[bv[i]], 1.0f);
}

__global__ void pool_sum(const float* __restrict__ h, const int* __restrict__ bv,
                         float* __restrict__ out) {
  const long long total = (long long)N_NODES * (EMB_DIM / 4);
  long long t = (long long)blockIdx.x * blockDim.x + threadIdx.x;
  if (t >= total) return;
  const int n = (int)(t >> 5);                  // EMB_DIM/4 == 32 chunks
  const int c = (int)(t & 31) << 2;
  const int g = bv[n];
  const float4 v = *(const float4*)(h + (long long)n * EMB_DIM + c);
  float* __restrict__ o = out + (long long)g * EMB_DIM + c;
  atomicAdd(o + 0, v.x);
  atomicAdd(o + 1, v.y);
  atomicAdd(o + 2, v.z);
  atomicAdd(o + 3, v.w);
}

__global__ void pool_div(float* __restrict__ out, const float* __restrict__ cnt) {
  int i = blockIdx.x * blockDim.x + threadIdx.x;
  if (i < N_GRAPHS * EMB_DIM) out[i] /= fmaxf(cnt[i >> 7], 1.0f);
}

// ------------------------------------------------------------------ driver

extern "C" void kernel_launch(void* const* d_in, const int* in_sizes, int n_in,
                              void* d_out, int out_size, void* d_ws, size_t ws_size,
                              hipStream_t stream) {
  (void)in_sizes; (void)n_in; (void)out_size; (void)ws_size;

  const float* x   = (const float*)d_in[0];
  const int*   ei  = (const int*)d_in[1];       // [2, E] flattened
  const float* ew  = (const float*)d_in[2];
  const int*   bv  = (const int*)d_in[3];
  const float* W1  = (const float*)d_in[4];
  const float* b1  = (const float*)d_in[5];
  const float* g1  = (const float*)d_in[6];
  const float* be1 = (const float*)d_in[7];
  const float* W2  = (const float*)d_in[8];
  const float* b2  = (const float*)d_in[9];
  const float* g2  = (const float*)d_in[10];
  const float* be2 = (const float*)d_in[11];
  float* out = (float*)d_out;

  const int* src = ei;
  const int* dst = ei + N_EDGES;

  // workspace carve-up (floats)
  float* ws   = (float*)d_ws;
  const long long N64  = (long long)N_NODES * HID_DIM;
  const long long N128 = (long long)N_NODES * EMB_DIM;
  float* h1   = ws;                 // N*64
  float* agg1 = h1 + N64;           // N*64
  float* h2   = agg1 + N64;         // N*128
  float* agg2 = h2 + N128;          // N*128
  float* ssum = agg2 + N128;        // 128
  float* sqs  = ssum + 128;         // 128
  float* mean = sqs + 128;          // 128
  float* rstd = mean + 128;         // 128
  float* cnt  = rstd + 128;         // 256

  const int TPB = 256;
  dim3 blk(TPB);

  // ---------------- layer 1: GCNConv(128 -> 64) ----------------
  init_bias<HID_DIM><<<(unsigned)((N64 + TPB - 1) / TPB), blk, 0, stream>>>(agg1, b1);
  gemm_wmma_f32<<<N_NODES / 16, (HID_DIM / 16) * 32, 0, stream>>>(x, W1, h1, HID_DIM, IN_DIM);
  {
    long long total = (long long)N_EDGES * (HID_DIM / 4);
    scatter_edges<HID_DIM><<<(unsigned)((total + TPB - 1) / TPB), blk, 0, stream>>>(
        h1, agg1, src, dst, ew);
  }
  zero_f<<<1, 256, 0, stream>>>(ssum, 256);   // ssum + sqs contiguous
  col_stats<HID_DIM><<<512, 256, 0, stream>>>(agg1, ssum, sqs);
  finalize_stats<<<1, 128, 0, stream>>>(ssum, sqs, mean, rstd, HID_DIM);
  bn_relu<HID_DIM><<<(unsigned)((N64 + TPB - 1) / TPB), blk, 0, stream>>>(
      agg1, h1, mean, rstd, g1, be1);          // h1 now holds relu(bn(agg1))

  // ---------------- layer 2: GCNConv(64 -> 128) ----------------
  init_bias<EMB_DIM><<<(unsigned)((N128 + TPB - 1) / TPB), blk, 0, stream>>>(agg2, b2);
  gemm_wmma_f32<<<N_NODES / 16, (EMB_DIM / 16) * 32, 0, stream>>>(h1, W2, h2, EMB_DIM, HID_DIM);
  {
    long long total = (long long)N_EDGES * (EMB_DIM / 4);
    scatter_edges<EMB_DIM><<<(unsigned)((total + TPB - 1) / TPB), blk, 0, stream>>>(
        h2, agg2, src, dst, ew);
  }
  zero_f<<<1, 256, 0, stream>>>(ssum, 256);
  col_stats<EMB_DIM><<<512, 256, 0, stream>>>(agg2, ssum, sqs);
  finalize_stats<<<1, 128, 0, stream>>>(ssum, sqs, mean, rstd, EMB_DIM);
  bn_relu<EMB_DIM><<<(unsigned)((N128 + TPB - 1) / TPB), blk, 0, stream>>>(
      agg2, h2, mean, rstd, g2, be2);          // h2 now holds relu(bn(agg2))

  // ---------------- global mean pool ----------------
  zero_f<<<(N_GRAPHS * EMB_DIM + TPB - 1) / TPB, blk, 0, stream>>>(out, N_GRAPHS * EMB_DIM);
  zero_f<<<1, 256, 0, stream>>>(cnt, N_GRAPHS);
  pool_count<<<(N_NODES + TPB - 1) / TPB, blk, 0, stream>>>(bv, cnt);
  {
    long long total = (long long)N_NODES * (EMB_DIM / 4);
    pool_sum<<<(unsigned)((total + TPB - 1) / TPB), blk, 0, stream>>>(h2, bv, out);
  }
  pool_div<<<(N_GRAPHS * EMB_DIM + TPB - 1) / TPB, blk, 0, stream>>>(out, cnt);
}